// SA_89300960019259
// MI455X (gfx1250) — compile-verified
//
#include <hip/hip_runtime.h>
#include <math.h>

// Problem shape (fixed by setup_inputs).
#define BB   32
#define CC   64
#define HH   128
#define WW2  128
#define HW   (HH * WW2)     // 16384
#define CR   38             // floor(0.6*64)=38, even
#define CIR  26

typedef __attribute__((ext_vector_type(2))) float v2f;
typedef __attribute__((ext_vector_type(4))) float v4f;
typedef __attribute__((ext_vector_type(8))) float v8f;

#if defined(__gfx1250__) && __has_builtin(__builtin_amdgcn_wmma_f32_16x16x4_f32)
#define HAVE_WMMA_F32F32 1
#else
#define HAVE_WMMA_F32F32 0
#endif

// ---------------------------------------------------------------------------
// Kernel 1: per-batch top-38 mask from M (exact jax.lax.top_k tie semantics:
// rank = #(value greater) + #(equal with smaller index); relevant if rank<38).
// ---------------------------------------------------------------------------
__global__ void sa_topk_mask(const float* __restrict__ M, float* __restrict__ mask) {
  __shared__ float mv[CC];
  const int b = blockIdx.x;
  const int c = threadIdx.x;
  const float v = M[b * CC + c];
  mv[c] = v;
  __syncthreads();
  int cnt = 0;
#pragma unroll
  for (int j = 0; j < CC; ++j) {
    const float u = mv[j];
    cnt += (u > v) || ((u == v) && (j < c));
  }
  mask[b * CC + c] = (cnt < CR) ? 1.0f : 0.0f;
}

// ---------------------------------------------------------------------------
// Kernel 2: per-pixel channel reductions.
// Each wave handles 16 pixels. Channel sums via chained V_WMMA_F32_16X16X4_F32:
//   A (16x4): row M=0 = r_mask for the 4 channels of this k-chunk,
//             row M=1 = 1.0 (total sum), rows 2..15 = 0.
//   B (4x16): x[b, k..k+3, pix..pix+15].
// A/B f32 lane layout (per ISA 7.12.2): lanes 0-15 hold K/N pair {0,1} in
// {v0,v1}; lanes 16-31 hold {2,3}. D: v0 lanes0-15 = row0 (r_sum),
// v1 lanes0-15 = row1 (total).  Maxes ride along in VALU on the same loads,
// combined across half-waves with shfl_xor(16) (wave32).
// ---------------------------------------------------------------------------
__global__ __launch_bounds__(256) void sa_reduce(
    const float* __restrict__ x, const float* __restrict__ mask,
    float* __restrict__ r_avg, float* __restrict__ r_max,
    float* __restrict__ ir_avg, float* __restrict__ ir_max) {
  __shared__ float mk[CC];
  const int b = blockIdx.y;
  if (threadIdx.x < CC) mk[threadIdx.x] = mask[b * CC + threadIdx.x];
  __syncthreads();

  const int wave = threadIdx.x >> 5;
  const int lane = threadIdx.x & 31;
  const int group = blockIdx.x * (blockDim.x >> 5) + wave;  // 16-pixel group
  const int pix = group * 16;
  const int n = lane & 15;                 // pixel / N index
  const int kk = (lane < 16) ? 0 : 2;      // K sub-row pair held by this lane
  const int mrow = lane & 15;              // M index for A-matrix lanes

  const float* xb = x + ((size_t)b * CC) * HW + (size_t)pix + n;

  v8f d = {0.f, 0.f, 0.f, 0.f, 0.f, 0.f, 0.f, 0.f};
  float sr = 0.f, st = 0.f;                // fallback sum accumulators
  float accr = 0.f, accir = 0.f;           // maxes (zeros are legitimately in the set)

#pragma unroll
  for (int k0 = 0; k0 < CC; k0 += 4) {
    const int c0 = k0 + kk;
    const float b0 = xb[(size_t)c0 * HW];
    const float b1 = xb[(size_t)(c0 + 1) * HW];
    const float m0 = mk[c0];
    const float m1 = mk[c0 + 1];
#if HAVE_WMMA_F32F32
    const float a0 = (mrow == 0) ? m0 : ((mrow == 1) ? 1.0f : 0.0f);
    const float a1 = (mrow == 0) ? m1 : ((mrow == 1) ? 1.0f : 0.0f);
    v2f Av = {a0, a1};
    v2f Bv = {b0, b1};
    d = __builtin_amdgcn_wmma_f32_16x16x4_f32(false, Av, false, Bv,
                                              (short)0, d, false, false);
#else
    sr += m0 * b0 + m1 * b1;
    st += b0 + b1;
#endif
    accr  = fmaxf(accr,  m0 * b0);
    accr  = fmaxf(accr,  m1 * b1);
    accir = fmaxf(accir, (1.0f - m0) * b0);
    accir = fmaxf(accir, (1.0f - m1) * b1);
  }

  float r_sum, tot;
#if HAVE_WMMA_F32F32
  r_sum = d[0];   // lanes 0-15: D[M=0][N=lane]
  tot   = d[1];   // lanes 0-15: D[M=1][N=lane]
#else
  sr += __shfl_xor(sr, 16, 32);
  st += __shfl_xor(st, 16, 32);
  r_sum = sr;
  tot = st;
#endif
  accr  = fmaxf(accr,  __shfl_xor(accr, 16, 32));
  accir = fmaxf(accir, __shfl_xor(accir, 16, 32));

  if (lane < 16) {
    const float scale_r  = (float)(64.0 / 38.0);  // C/C_r cast to f32, as reference
    const float scale_ir = (float)(64.0 / 26.0);
    const size_t idx = (size_t)b * HW + pix + lane;
    const float ir_sum = tot - r_sum;
    r_avg[idx]  = (r_sum  * (1.0f / 64.0f)) * scale_r;   // mean * (C/C_r)
    ir_avg[idx] = (ir_sum * (1.0f / 64.0f)) * scale_ir;
    r_max[idx]  = accr;
    ir_max[idx] = accir;
  }
}

// ---------------------------------------------------------------------------
// Kernel 3: 2-in/1-out 7x7 "same" conv + BN + sigmoid(relu(.)) for both the
// relevant (A_S1) and irrelevant (A_S2) attention maps. Weights in LDS.
// ---------------------------------------------------------------------------
__global__ __launch_bounds__(256) void sa_conv_act(
    const float* __restrict__ ravg, const float* __restrict__ rmax,
    const float* __restrict__ iravg, const float* __restrict__ irmax,
    const float* __restrict__ wgt,  // (1,2,7,7) flat: [i*49 + kh*7 + kw]
    const float* __restrict__ g, const float* __restrict__ be,
    const float* __restrict__ mu, const float* __restrict__ var,
    float* __restrict__ A1, float* __restrict__ A2) {
  __shared__ float wk[98];
  if (threadIdx.x < 98) wk[threadIdx.x] = wgt[threadIdx.x];
  __syncthreads();

  const int idx = blockIdx.x * blockDim.x + threadIdx.x;
  if (idx >= BB * HW) return;
  const int b = idx >> 14;             // / HW
  const int p = idx & (HW - 1);
  const int h = p >> 7;
  const int w = p & 127;

  const float scale = g[0] * rsqrtf(var[0] + 1e-5f);
  const float shift = be[0] - mu[0] * scale;

  const size_t bo = (size_t)b * HW;
  float acc1 = 0.f, acc2 = 0.f;
#pragma unroll
  for (int kh = 0; kh < 7; ++kh) {
    const int hh = h + kh - 3;
    if (hh < 0 || hh >= HH) continue;
#pragma unroll
    for (int kw = 0; kw < 7; ++kw) {
      const int ww = w + kw - 3;
      if (ww < 0 || ww >= WW2) continue;
      const size_t off = bo + (size_t)hh * WW2 + ww;
      const float w0 = wk[kh * 7 + kw];
      const float w1 = wk[49 + kh * 7 + kw];
      acc1 = fmaf(w0, ravg[off],  fmaf(w1, rmax[off],  acc1));
      acc2 = fmaf(w0, iravg[off], fmaf(w1, irmax[off], acc2));
    }
  }
  const float z1 = fmaxf(fmaf(acc1, scale, shift), 0.0f);
  const float z2 = fmaxf(fmaf(acc2, scale, shift), 0.0f);
  A1[idx] = 1.0f / (1.0f + expf(-z1));
  A2[idx] = 1.0f / (1.0f + expf(-z2));
}

// ---------------------------------------------------------------------------
// Kernel 4: out = x * (mask ? A_S1 : A_S2), 128-bit vectorized. Output stored
// non-temporally so the 128 MiB stream does not evict x from the 192 MB L2
// (x was warmed by kernel 2, so the re-read here should largely hit L2).
// Uses ext_vector float4 (v4f) — HIP's float4 class is rejected by
// __builtin_nontemporal_store.
// ---------------------------------------------------------------------------
__global__ __launch_bounds__(256) void sa_combine(
    const float* __restrict__ x, const float* __restrict__ mask,
    const float* __restrict__ A1, const float* __restrict__ A2,
    float* __restrict__ out) {
  const size_t i4 = (size_t)blockIdx.x * blockDim.x + threadIdx.x;
  const size_t N4 = (size_t)BB * CC * HW / 4;
  if (i4 >= N4) return;
  const size_t e = i4 * 4;
  const int p  = (int)(e & (HW - 1));
  const int bc = (int)(e >> 14);        // b*CC + c
  const int b  = bc >> 6;

  const v4f xv = ((const v4f*)x)[i4];
  const float* Am = (mask[bc] > 0.5f) ? A1 : A2;
  const v4f av = *(const v4f*)(Am + (size_t)b * HW + p);
  const v4f o = xv * av;
  __builtin_nontemporal_store(o, (v4f*)out + i4);
}

// ---------------------------------------------------------------------------
extern "C" void kernel_launch(void* const* d_in, const int* in_sizes, int n_in,
                              void* d_out, int out_size, void* d_ws, size_t ws_size,
                              hipStream_t stream) {
  const float* x      = (const float*)d_in[0];
  const float* M      = (const float*)d_in[1];
  const float* conv_w = (const float*)d_in[2];
  const float* bn_g   = (const float*)d_in[3];
  const float* bn_b   = (const float*)d_in[4];
  const float* bn_mu  = (const float*)d_in[5];
  const float* bn_var = (const float*)d_in[6];
  float* out = (float*)d_out;

  // Workspace layout (floats), 16B-aligned chunks.
  float* ws     = (float*)d_ws;
  float* mask   = ws;                        // 2048
  float* r_avg  = ws + 2048;                 // 524288 each below
  float* r_max  = r_avg  + (size_t)BB * HW;
  float* ir_avg = r_max  + (size_t)BB * HW;
  float* ir_max = ir_avg + (size_t)BB * HW;
  float* A1     = ir_max + (size_t)BB * HW;
  float* A2     = A1     + (size_t)BB * HW;

  // K1: top-k mask. One 64-thread block per batch.
  sa_topk_mask<<<BB, CC, 0, stream>>>(M, mask);

  // K2: channel reductions. 8 waves/block, 1 wave = 16 pixels,
  // 16384 px/batch = 1024 groups -> 128 blocks.x per batch.
  dim3 g2(HW / 16 / 8, BB);
  sa_reduce<<<g2, 256, 0, stream>>>(x, mask, r_avg, r_max, ir_avg, ir_max);

  // K3: conv + BN + sigmoid(relu).
  sa_conv_act<<<(BB * HW + 255) / 256, 256, 0, stream>>>(
      r_avg, r_max, ir_avg, ir_max, conv_w, bn_g, bn_b, bn_mu, bn_var, A1, A2);

  // K4: final combine, 128-bit vectorized.
  const size_t N4 = (size_t)BB * CC * HW / 4;
  sa_combine<<<(unsigned)((N4 + 255) / 256), 256, 0, stream>>>(x, mask, A1, A2, out);
}